// HashEmbedding_54778012893289
// MI455X (gfx1250) — compile-verified
//
#include <hip/hip_runtime.h>

// Multiresolution hash-grid embedding (InstantNGP-style) for gfx1250.
//
// Layout decisions (MI455X-specific):
//  - tables (64 MB) < L2 (192 MB): keep them L2-resident; protect with NT
//    output stores (th:NT) so the 128 MB streaming write doesn't evict them.
//  - thread = (point b, level l) with l = tid & 15: a wave's 32 x 8 B output
//    stores form one contiguous 256 B span -> fully coalesced; the 12 B x[b]
//    read is shared by 16 consecutive lanes (cache broadcast).
//  - warm kernel issues global_prefetch_b8 over the table to populate L2
//    ahead of the 128M-gather storm.

typedef __attribute__((ext_vector_type(2))) float v2f;

namespace {
constexpr int      kLevels = 16;
constexpr int      kLog2T  = 19;
constexpr unsigned kMask   = (1u << kLog2T) - 1u;
constexpr unsigned kP1     = 2654435761u;
constexpr unsigned kP2     = 805459861u;
constexpr int      kBsz    = 1 << 20;
}  // namespace

// floor(16 * 2^(i/3)) for i = 0..15 (matches numpy float64 floor).
__constant__ int kResTab[kLevels] = {16, 20, 25, 32, 40, 50, 64, 80,
                                     101, 128, 161, 203, 256, 322, 406, 512};

__global__ __launch_bounds__(256) void warm_l2_kernel(
    const char* __restrict__ tables, unsigned nbytes) {
  unsigned i = (blockIdx.x * 256u + threadIdx.x) * 512u;  // 4 x 128B lines/thread
  if (i < nbytes) {
    __builtin_prefetch(tables + i + 0, 0, 1);
    __builtin_prefetch(tables + i + 128, 0, 1);
    __builtin_prefetch(tables + i + 256, 0, 1);
    __builtin_prefetch(tables + i + 384, 0, 1);
  }
}

__global__ __launch_bounds__(256) void hash_embed_kernel(
    const float* __restrict__ x, const float* __restrict__ tables,
    float* __restrict__ out) {
  const unsigned tid = blockIdx.x * 256u + threadIdx.x;
  const unsigned lvl = tid & (unsigned)(kLevels - 1);
  const unsigned b   = tid >> 4;

  // point coords, shared by 16 consecutive lanes (one line per half-wave)
  const float px = x[3u * b + 0];
  const float py = x[3u * b + 1];
  const float pz = x[3u * b + 2];

  const int   res  = kResTab[lvl];
  const float hres = 0.5f * (float)res;          // u = (x+1)*res/2
  const float rmax = (float)(res - 1);

  const float ux = fmaf(px, hres, hres);
  const float uy = fmaf(py, hres, hres);
  const float uz = fmaf(pz, hres, hres);

  const float fx = fminf(fmaxf(floorf(ux), 0.0f), rmax);
  const float fy = fminf(fmaxf(floorf(uy), 0.0f), rmax);
  const float fz = fminf(fmaxf(floorf(uz), 0.0f), rmax);

  const float tx = ux - fx, ty = uy - fy, tz = uz - fz;

  const unsigned ix = (unsigned)fx;
  const unsigned iy = (unsigned)fy;
  const unsigned iz = (unsigned)fz;

  // spatial hash: h = (x*1 ^ y*P1 ^ z*P2) & mask ; precompute both offsets/dim
  const unsigned hx0 = ix,              hx1 = ix + 1u;
  const unsigned hy0 = iy * kP1,        hy1 = (iy + 1u) * kP1;
  const unsigned hz0 = iz * kP2,        hz1 = (iz + 1u) * kP2;

  const v2f* __restrict__ tab =
      reinterpret_cast<const v2f*>(tables) + ((size_t)lvl << kLog2T);

  // corner c: bit0 -> x offset, bit1 -> y, bit2 -> z. Issue all 8 gathers
  // back-to-back so they share one loadcnt wait (L2-resident hits).
  const v2f e0 = tab[(hx0 ^ hy0 ^ hz0) & kMask];
  const v2f e1 = tab[(hx1 ^ hy0 ^ hz0) & kMask];
  const v2f e2 = tab[(hx0 ^ hy1 ^ hz0) & kMask];
  const v2f e3 = tab[(hx1 ^ hy1 ^ hz0) & kMask];
  const v2f e4 = tab[(hx0 ^ hy0 ^ hz1) & kMask];
  const v2f e5 = tab[(hx1 ^ hy0 ^ hz1) & kMask];
  const v2f e6 = tab[(hx0 ^ hy1 ^ hz1) & kMask];
  const v2f e7 = tab[(hx1 ^ hy1 ^ hz1) & kMask];

  // trilinear weights
  const float sx = 1.0f - tx, sy = 1.0f - ty, sz = 1.0f - tz;
  const float w00 = sx * sy, w10 = tx * sy, w01 = sx * ty, w11 = tx * ty;
  const float c0 = w00 * sz, c1 = w10 * sz, c2 = w01 * sz, c3 = w11 * sz;
  const float c4 = w00 * tz, c5 = w10 * tz, c6 = w01 * tz, c7 = w11 * tz;

  float r0 = c0 * e0.x, r1 = c0 * e0.y;
  r0 = fmaf(c1, e1.x, r0); r1 = fmaf(c1, e1.y, r1);
  r0 = fmaf(c2, e2.x, r0); r1 = fmaf(c2, e2.y, r1);
  r0 = fmaf(c3, e3.x, r0); r1 = fmaf(c3, e3.y, r1);
  r0 = fmaf(c4, e4.x, r0); r1 = fmaf(c4, e4.y, r1);
  r0 = fmaf(c5, e5.x, r0); r1 = fmaf(c5, e5.y, r1);
  r0 = fmaf(c6, e6.x, r0); r1 = fmaf(c6, e6.y, r1);
  r0 = fmaf(c7, e7.x, r0); r1 = fmaf(c7, e7.y, r1);

  v2f r;
  r.x = r0;
  r.y = r1;
  // streaming 128 MB output: non-temporal so it can't evict the L2-resident
  // tables. Wave-contiguous: lanes cover [b0*32, b0*32+64) floats.
  v2f* __restrict__ o = reinterpret_cast<v2f*>(out);
  __builtin_nontemporal_store(r, o + ((size_t)b * kLevels + lvl));
}

extern "C" void kernel_launch(void* const* d_in, const int* in_sizes, int n_in,
                              void* d_out, int out_size, void* d_ws, size_t ws_size,
                              hipStream_t stream) {
  (void)in_sizes; (void)n_in; (void)out_size; (void)d_ws; (void)ws_size;
  const float* x      = (const float*)d_in[0];
  const float* tables = (const float*)d_in[1];
  float*       out    = (float*)d_out;

  // Warm the 64 MB hash table into L2 (prefetch path; near-free if resident).
  const unsigned tbytes = (unsigned)kLevels * (1u << kLog2T) * 2u * 4u;  // 64 MiB
  const unsigned warm_threads = tbytes / 512u;                           // 131072
  warm_l2_kernel<<<warm_threads / 256u, 256, 0, stream>>>(
      (const char*)tables, tbytes);

  const unsigned total = (unsigned)kBsz * (unsigned)kLevels;  // 16.7M threads
  hash_embed_kernel<<<total / 256u, 256, 0, stream>>>(x, tables, out);
}